// Longformer_biLSTM_CRF_40870908789305
// MI455X (gfx1250) — compile-verified
//
#include <hip/hip_runtime.h>
#include <stdint.h>

#define B 128
#define T 4096
#define L 64
#define START_ID 62
#define NEG_INF -10000.0f
#define CHUNK 64                 // timesteps per TDM tile: 64*64*4B = 16KB
#define NCHUNK (T / CHUNK)

typedef unsigned int v4u __attribute__((ext_vector_type(4)));
typedef int          v8i __attribute__((ext_vector_type(8)));
typedef int          v4i __attribute__((ext_vector_type(4)));

#if defined(__has_builtin)
#if __has_builtin(__builtin_amdgcn_tensor_load_to_lds) && __has_builtin(__builtin_amdgcn_s_wait_tensorcnt)
#define HAVE_TDM 1
#endif
#endif

// Issue one TDM descriptor: contiguous 1-D copy of `nelem` f32 from global -> LDS.
__device__ __forceinline__ void tdm_load_chunk(const float* gsrc, unsigned lds_byte_addr, int nelem) {
#ifdef HAVE_TDM
  unsigned long long ga = (unsigned long long)(const void*)gsrc;
  unsigned n = (unsigned)nelem;
  v4u g0;
  g0.x = 1u;                                                   // count=1 (valid user descriptor)
  g0.y = lds_byte_addr;                                        // lds_addr [63:32]
  g0.z = (unsigned)(ga & 0xFFFFFFFFull);                       // global_addr lo
  g0.w = (unsigned)((ga >> 32) & 0x01FFFFFFull) | (2u << 30);  // global_addr hi | type=2
  v8i g1 = {0, 0, 0, 0, 0, 0, 0, 0};
  g1[0] = (int)(2u << 16);                                     // data_size = 4 bytes
  g1[1] = (int)((n & 0xFFFFu) << 16);                          // tensor_dim0 [63:48]
  g1[2] = (int)(((n >> 16) & 0xFFFFu) | (1u << 16));           // tensor_dim0 hi | tensor_dim1=1
  g1[3] = (int)((n & 0xFFFFu) << 16);                          // tile_dim0 [127:112]
  g1[4] = 1;                                                   // tile_dim1 = 1
  g1[5] = (int)n;                                              // tensor_dim0_stride lo32
  g1[6] = (int)((n & 0xFFFFu) << 16);                          // tensor_dim1_stride lo16
  v4i gz = {0, 0, 0, 0};
#if defined(__clang_major__) && (__clang_major__ >= 23)
  v8i gz8 = {0, 0, 0, 0, 0, 0, 0, 0};
  __builtin_amdgcn_tensor_load_to_lds(g0, g1, gz, gz, gz8, 0);
#else
  __builtin_amdgcn_tensor_load_to_lds(g0, g1, gz, gz, 0);
#endif
#else
  (void)gsrc; (void)lds_byte_addr; (void)nelem;
#endif
}

// One wave32 per batch. Lane `tid` owns labels tid and tid+32.
// delta lives in registers; cross-lane broadcast via v_readlane (wave32).
__global__ void __launch_bounds__(32) viterbi_forward(const float* __restrict__ feats,
                                                      const float* __restrict__ trans,
                                                      float* __restrict__ out_score,
                                                      unsigned char* __restrict__ psi,
                                                      int* __restrict__ last_lab) {
  const int tid = threadIdx.x;
  const int b = blockIdx.x;
  __shared__ float featbuf[2][CHUNK * L];   // 32 KB double buffer

  // Transition rows for this lane's two labels -> 128 VGPRs.
  float tr0[L], tr1[L];
  const float4* rA = (const float4*)(trans + (size_t)tid * L);
  const float4* rB = (const float4*)(trans + (size_t)(tid + 32) * L);
#pragma unroll
  for (int k = 0; k < 16; ++k) {
    float4 a = rA[k]; float4 c = rB[k];
    tr0[4*k+0]=a.x; tr0[4*k+1]=a.y; tr0[4*k+2]=a.z; tr0[4*k+3]=a.w;
    tr1[4*k+0]=c.x; tr1[4*k+1]=c.y; tr1[4*k+2]=c.z; tr1[4*k+3]=c.w;
  }

  float delta0 = NEG_INF;                                  // labels 0..31
  float delta1 = (tid + 32 == START_ID) ? 0.0f : NEG_INF;  // labels 32..63

  const float* fb = feats + (size_t)b * T * L;

#ifdef HAVE_TDM
  tdm_load_chunk(fb, (unsigned)(size_t)(void*)&featbuf[0][0], CHUNK * L);
#endif

#pragma unroll 1
  for (int c = 0; c < NCHUNK; ++c) {
    const int sel = c & 1;
#ifdef HAVE_TDM
    if (c + 1 < NCHUNK) {
      tdm_load_chunk(fb + (size_t)(c + 1) * CHUNK * L,
                     (unsigned)(size_t)(void*)&featbuf[sel ^ 1][0], CHUNK * L);
      __builtin_amdgcn_s_wait_tensorcnt(1);   // chunk c complete (in-order per wave)
    } else {
      __builtin_amdgcn_s_wait_tensorcnt(0);
    }
#else
    for (int k = tid; k < CHUNK * L; k += 32)
      featbuf[sel][k] = fb[(size_t)c * CHUNK * L + k];
    __syncthreads();
#endif

    const int t0 = (c == 0) ? 1 : 0;          // feats[0] unused by the recurrence
#pragma unroll 1
    for (int tt = t0; tt < CHUNK; ++tt) {
      const int t = c * CHUNK + tt;
      // 8 independent accumulation groups per label -> 16 parallel chains,
      // then a depth-3 tree merge (keeps first-occurrence argmax semantics).
      float bb0[8], bb1[8]; int aa0[8], aa1[8];
#pragma unroll
      for (int g = 0; g < 8; ++g) {
        bb0[g] = -__builtin_inff(); aa0[g] = 0;
        bb1[g] = -__builtin_inff(); aa1[g] = 0;
      }
#pragma unroll
      for (int j = 0; j < 64; ++j) {
        const int dsrc = (j < 32) ? __float_as_int(delta0) : __float_as_int(delta1);
        const float dj = __int_as_float(__builtin_amdgcn_readlane(dsrc, j & 31));
        const int g = j >> 3;
        float s0 = tr0[j] + dj; if (s0 > bb0[g]) { bb0[g] = s0; aa0[g] = j; }
        float s1 = tr1[j] + dj; if (s1 > bb1[g]) { bb1[g] = s1; aa1[g] = j; }
      }
#pragma unroll
      for (int off = 1; off < 8; off <<= 1) {
#pragma unroll
        for (int g = 0; g < 8; g += 2 * off) {
          if (bb0[g + off] > bb0[g]) { bb0[g] = bb0[g + off]; aa0[g] = aa0[g + off]; }
          if (bb1[g + off] > bb1[g]) { bb1[g] = bb1[g + off]; aa1[g] = aa1[g + off]; }
        }
      }
      unsigned char* p = psi + ((size_t)(t - 1) * B + b) * L;
      p[tid]      = (unsigned char)aa0[0];
      p[tid + 32] = (unsigned char)aa1[0];
      const float f0 = featbuf[sel][tt * L + tid];
      const float f1 = featbuf[sel][tt * L + tid + 32];
      delta0 = bb0[0] + f0;
      delta1 = bb1[0] + f1;
    }
  }

  // Final score + argmax over labels 0..63 (ascending, first occurrence).
  float best = -__builtin_inff(); int lab = 0;
#pragma unroll
  for (int j = 0; j < 32; ++j) {
    float v = __int_as_float(__builtin_amdgcn_readlane(__float_as_int(delta0), j));
    if (v > best) { best = v; lab = j; }
  }
#pragma unroll
  for (int j = 0; j < 32; ++j) {
    float v = __int_as_float(__builtin_amdgcn_readlane(__float_as_int(delta1), j));
    if (v > best) { best = v; lab = 32 + j; }
  }
  if (tid == 0) { out_score[b] = best; last_lab[b] = lab; }
}

// 128 independent backtrack chains; psi (33.5 MB) is L2-resident (192 MB L2).
// The dependent byte-load chain is hidden by prefetching the known psi row
// 16 steps ahead (global_prefetch_b8).
__global__ void viterbi_backtrack(const unsigned char* __restrict__ psi,
                                  const int* __restrict__ last_lab,
                                  float* __restrict__ out_path) {
  int b = blockIdx.x * blockDim.x + threadIdx.x;
  if (b >= B) return;
  int cur = last_lab[b];
  float* row = out_path + (size_t)b * T;
  row[T - 1] = (float)cur;
#pragma unroll 1
  for (int t = T - 2; t >= 0; --t) {
    if (t >= 16) __builtin_prefetch(psi + ((size_t)(t - 16) * B + b) * L, 0, 0);
    cur = psi[((size_t)t * B + b) * L + cur];
    row[t] = (float)cur;
  }
}

extern "C" void kernel_launch(void* const* d_in, const int* in_sizes, int n_in,
                              void* d_out, int out_size, void* d_ws, size_t ws_size,
                              hipStream_t stream) {
  (void)in_sizes; (void)n_in; (void)out_size; (void)ws_size;
  const float* feats = (const float*)d_in[0];        // [B, T, L] f32
  const float* trans = (const float*)d_in[1];        // [L, L] f32
  float* out_score = (float*)d_out;                  // [B]
  float* out_path  = out_score + B;                  // [B, T] (labels as float)
  unsigned char* psi = (unsigned char*)d_ws;         // [(T-1), B, L] backpointers
  int* last_lab = (int*)(psi + (size_t)(T - 1) * B * L);

  viterbi_forward<<<B, 32, 0, stream>>>(feats, trans, out_score, psi, last_lab);
  viterbi_backtrack<<<(B + 31) / 32, 32, 0, stream>>>(psi, last_lab, out_path);
}